// Layer_63041529970737
// MI455X (gfx1250) — compile-verified
//
#include <hip/hip_runtime.h>

// ---------------- problem constants ----------------
#define NNODES 10000
#define NEDGES 160000

typedef __attribute__((ext_vector_type(2))) float v2f;
typedef __attribute__((ext_vector_type(8))) float v8f;

// workspace float offsets
constexpr size_t OFF_A   = 0;          // 10000 * 128
constexpr size_t OFF_B   = 1280000;    // 10000 * 32
constexpr size_t OFF_C   = 1600000;    // 10000 * 96
constexpr size_t OFF_CNT = 2560000;    // 10000
constexpr size_t OFF_W0  = 2570000;    // 64*32
constexpr size_t OFF_W1  = 2572048;    // 64*32
constexpr size_t ZERO_FLOATS = 2570000; // A..cnt region (W written by prep)

// ---------------- zero accumulators ----------------
__global__ void zero_ws(float* __restrict__ ws) {
    size_t base = ((size_t)blockIdx.x * blockDim.x + threadIdx.x) * 4;
    if (base < ZERO_FLOATS) {
        float4 z = make_float4(0.f, 0.f, 0.f, 0.f);
        *(float4*)(ws + base) = z;
    }
}

// ---------------- fold lin + shortcut weights ----------------
// W0 = [lin_w0[:32]/12 + sc_w0/sqrt(32) ; lin_w0[32:]/(12*sqrt(3))]
// W1 = [lin_w1[:32]/12 + sc_w1/sqrt(32) ; lin_w1[32:]/12]
__global__ void prep_weights(const float* __restrict__ lw0, const float* __restrict__ lw1,
                             const float* __restrict__ sw0, const float* __restrict__ sw1,
                             float* __restrict__ ws) {
    int t = blockIdx.x * blockDim.x + threadIdx.x;   // 0 .. 2047
    if (t >= 64 * 32) return;
    int k = t >> 5;
    const float c_lin  = 1.0f / 12.0f;                       // 1/(1.5 * sqrt(64))
    const float c_sc   = 0.17677669529663687f;               // 1/sqrt(32)
    const float c_b0   = 1.0f / (12.0f * 1.7320508075688772f); // /sqrt(3) folded
    float w0, w1;
    if (k < 32) {
        w0 = lw0[t] * c_lin + sw0[t] * c_sc;
        w1 = lw1[t] * c_lin + sw1[t] * c_sc;
    } else {
        w0 = lw0[t] * c_b0;
        w1 = lw1[t] * c_lin;
    }
    ws[OFF_W0 + t] = w0;
    ws[OFF_W1 + t] = w1;
}

// ---------------- edge scatter-accumulate ----------------
// one wave per edge; lane = channel u (0..31); coalesced 128B loads of nf[sender]
// A[r,128] += nf ; B[r,u] += sum_i sh_i * f1[i][u] ; C[r,i,u] += sh_i * f0[u]
__global__ void edge_kernel(const float* __restrict__ nf, const float* __restrict__ sh,
                            const int* __restrict__ senders, const int* __restrict__ receivers,
                            float* __restrict__ ws) {
    int wid  = threadIdx.x >> 5;
    int lane = threadIdx.x & 31;
    int e = blockIdx.x * 8 + wid;
    if (e >= NEDGES) return;

    int s = senders[e];
    int r = receivers[e];
    float sh0 = sh[e * 3 + 0], sh1 = sh[e * 3 + 1], sh2 = sh[e * 3 + 2];

    const float* nfp = nf + (size_t)s * 128;
    float f0  = nfp[lane];
    float f1a = nfp[32 + lane];
    float f1b = nfp[64 + lane];
    float f1c = nfp[96 + lane];

    float* A = ws + OFF_A + (size_t)r * 128;
    unsafeAtomicAdd(A + lane,      f0);
    unsafeAtomicAdd(A + 32 + lane, f1a);
    unsafeAtomicAdd(A + 64 + lane, f1b);
    unsafeAtomicAdd(A + 96 + lane, f1c);

    float bsum = sh0 * f1a + sh1 * f1b + sh2 * f1c;   // /sqrt(3) folded into W0
    unsafeAtomicAdd(ws + OFF_B + (size_t)r * 32 + lane, bsum);

    float* C = ws + OFF_C + (size_t)r * 96;
    unsafeAtomicAdd(C + lane,      sh0 * f0);
    unsafeAtomicAdd(C + 32 + lane, sh1 * f0);
    unsafeAtomicAdd(C + 64 + lane, sh2 * f0);

    if (lane == 0) unsafeAtomicAdd(ws + OFF_CNT + r, 1.0f);
}

// ---------------- node GEMM with V_WMMA_F32_16X16X4_F32 ----------------
// 2500 row-tiles of 16: tiles [0,625) -> scalar rows X=[A0|B] with W0,
// tiles [625,2500) -> vector rows X=[A1_i|C_i] with W1. One wave per tile,
// two 16x16 N-tiles (N=32), K=64 in 16 steps of 4. Output scaled by 1/max(cnt,1).
__global__ void node_gemm(float* __restrict__ out, const float* __restrict__ ws) {
    int wid  = threadIdx.x >> 5;
    int lane = threadIdx.x & 31;
    int tile = blockIdx.x * 8 + wid;
    if (tile >= 2500) return;            // uniform per wave: EXEC stays all-ones

    int half = lane >> 4;                // 0: K0/K1 & D rows 0-7 ; 1: K2/K3 & D rows 8-15 halves
    int ml   = lane & 15;

    bool isV = tile >= 625;
    int rowbase = isV ? (tile - 625) * 16 : tile * 16;
    const float* W = ws + (isV ? OFF_W1 : OFF_W0);   // 64x32 row-major

    // per-lane X row pointers (lane ml owns row rowbase+ml for A-fragment loads)
    int myrow = rowbase + ml;
    const float *x_lo, *x_hi;
    if (!isV) {
        x_lo = ws + OFF_A + (size_t)myrow * 128;             // A0[n,k]
        x_hi = ws + OFF_B + (size_t)myrow * 32;              // B[n,k-32]
    } else {
        int n = myrow / 3, i = myrow - n * 3;
        x_lo = ws + OFF_A + (size_t)n * 128 + 32 + i * 32;   // A1[n,i,k]
        x_hi = ws + OFF_C + (size_t)n * 96 + i * 32;         // C[n,i,k-32]
    }

    v8f c0 = {};
    v8f c1 = {};
    #pragma unroll
    for (int k0 = 0; k0 < 64; k0 += 4) {
        int ka = k0 + 2 * half;          // even, so (ka, ka+1) never straddles 32
        float ax = (ka < 32) ? x_lo[ka]     : x_hi[ka - 32];
        float ay = (ka < 32) ? x_lo[ka + 1] : x_hi[ka + 1 - 32];
        v2f a  = {ax, ay};
        v2f b0 = { W[ka * 32 + ml],      W[(ka + 1) * 32 + ml] };
        v2f b1 = { W[ka * 32 + 16 + ml], W[(ka + 1) * 32 + 16 + ml] };
        c0 = __builtin_amdgcn_wmma_f32_16x16x4_f32(false, a, false, b0, (short)0, c0, false, false);
        c1 = __builtin_amdgcn_wmma_f32_16x16x4_f32(false, a, false, b1, (short)0, c1, false, false);
    }

    const float* cnt = ws + OFF_CNT;
    #pragma unroll
    for (int rr = 0; rr < 8; ++rr) {
        int row = rowbase + rr + 8 * half;   // D layout: VGPR rr -> M=rr (half 0) / M=8+rr (half 1)
        int n, colbase;
        if (!isV) { n = row; colbase = 0; }
        else      { n = row / 3; int i = row - n * 3; colbase = 32 + i * 32; }
        float m = 1.0f / fmaxf(cnt[n], 1.0f);
        out[(size_t)n * 128 + colbase + ml]      = c0[rr] * m;
        out[(size_t)n * 128 + colbase + 16 + ml] = c1[rr] * m;
    }
}

// ---------------- launcher ----------------
extern "C" void kernel_launch(void* const* d_in, const int* in_sizes, int n_in,
                              void* d_out, int out_size, void* d_ws, size_t ws_size,
                              hipStream_t stream) {
    const float* nf   = (const float*)d_in[0];   // (10000, 128)
    const float* sh   = (const float*)d_in[1];   // (160000, 3)
    const float* lw0  = (const float*)d_in[2];   // (64, 32)
    const float* lw1  = (const float*)d_in[3];   // (64, 32)
    const float* sw0  = (const float*)d_in[4];   // (32, 32)
    const float* sw1  = (const float*)d_in[5];   // (32, 32)
    const int*   snd  = (const int*)d_in[6];     // (160000,)
    const int*   rcv  = (const int*)d_in[7];     // (160000,)
    float* out = (float*)d_out;                  // (10000, 128)
    float* ws  = (float*)d_ws;

    // 1) zero accumulators (642500 float4 stores)
    zero_ws<<<(642500 + 255) / 256, 256, 0, stream>>>(ws);
    // 2) fold weights
    prep_weights<<<8, 256, 0, stream>>>(lw0, lw1, sw0, sw1, ws);
    // 3) edge scatter (8 edges per 256-thread block)
    edge_kernel<<<NEDGES / 8, 256, 0, stream>>>(nf, sh, snd, rcv, ws);
    // 4) node GEMM via f32 WMMA (2500 tiles, 8 waves per block)
    node_gemm<<<(2500 + 7) / 8, 256, 0, stream>>>(out, ws);
}